// GCNLayer_51565377356512
// MI455X (gfx1250) — compile-verified
//
#include <hip/hip_runtime.h>
#include <hip/hip_bf16.h>

// ---------------------------------------------------------------------------
// GCN layer for MI455X (gfx1250):
//   hw  = (h @ W) * norm          -> WMMA f32 16x16x4 (full fp32 precision)
//   agg = segment_sum(hw[src],dst)-> float4 gather + global_atomic_add_f32
//   out = relu(agg * norm + bias) -> elementwise finalize
// ---------------------------------------------------------------------------

typedef __attribute__((ext_vector_type(2))) float v2f;
typedef __attribute__((ext_vector_type(8))) float v8f;

#define DIN   128
#define DOUT  128
#define ROW_TILES_PER_BLOCK 5      // 625 blocks * 5 tiles * 16 rows = 50000
#define HPAD  132                  // padded LDS row stride (bank-conflict free)

// --------------------------------------------------------------------- zero
__global__ __launch_bounds__(256)
void gcn_zero_kernel(float4* __restrict__ out, int n4) {
    int gid = blockIdx.x * 256 + threadIdx.x;
    if (gid < n4) out[gid] = make_float4(0.f, 0.f, 0.f, 0.f);
}

// ------------------------------------------------------- WMMA GEMM + norm
// Block: 256 threads = 8 waves. Wave w owns column tile [w*16, w*16+16).
// Each block processes ROW_TILES_PER_BLOCK row tiles of 16 rows; the 16x128
// h tile is staged once into LDS and shared by all 8 waves.
__global__ __launch_bounds__(256)
void gcn_gemm_norm_kernel(const float* __restrict__ h,
                          const float* __restrict__ w,
                          const float* __restrict__ norm,
                          float* __restrict__ hw) {
    __shared__ float hlds[16 * HPAD];

    const int tid     = threadIdx.x;
    const int lane    = tid & 31;
    const int wave    = tid >> 5;        // 0..7  -> column tile
    const int l16     = lane & 15;
    const int half    = lane >> 4;       // 0/1   -> K split (ISA A layout)
    const int colBase = wave * 16;

    // Pull the 64KB weight matrix toward the caches (global_prefetch_b8).
    __builtin_prefetch(w + (size_t)tid * 64, 0, 1);

    for (int t = 0; t < ROW_TILES_PER_BLOCK; ++t) {
        const int rowBase = (blockIdx.x * ROW_TILES_PER_BLOCK + t) * 16;

        // Cooperative stage of the 16x128 h tile (2048 floats, float4 loads).
        #pragma unroll
        for (int it = 0; it < 2; ++it) {
            int f4 = tid + it * 256;          // 0..511 float4 slots
            int r  = f4 >> 5;                 // 32 float4 per row
            int c  = (f4 & 31) * 4;
            float4 v = *(const float4*)(h + (size_t)(rowBase + r) * DIN + c);
            hlds[r * HPAD + c + 0] = v.x;
            hlds[r * HPAD + c + 1] = v.y;
            hlds[r * HPAD + c + 2] = v.z;
            hlds[r * HPAD + c + 3] = v.w;
        }
        __syncthreads();

        // K-loop: 32 x V_WMMA_F32_16X16X4_F32.
        // A (16x4): lanes 0-15 hold K=k+0,k+1 ; lanes 16-31 hold K=k+2,k+3.
        // B (4x16): row K striped across lanes within a VGPR, mirrored halves.
        v8f acc = {};
        #pragma unroll 8
        for (int k = 0; k < DIN; k += 4) {
            const int kk = k + half * 2;
            v2f a;
            a.x = hlds[l16 * HPAD + kk + 0];
            a.y = hlds[l16 * HPAD + kk + 1];
            v2f b;
            b.x = w[(size_t)(kk + 0) * DOUT + colBase + l16];
            b.y = w[(size_t)(kk + 1) * DOUT + colBase + l16];
            acc = __builtin_amdgcn_wmma_f32_16x16x4_f32(
                false, a, false, b, (short)0, acc, false, false);
        }

        // C/D layout: VGPR v holds row M = v (lanes 0-15) or v+8 (lanes 16-31),
        // column N = lane%16. Apply pre-aggregation norm scaling and store.
        #pragma unroll
        for (int v = 0; v < 8; ++v) {
            const int m   = v + half * 8;
            const int row = rowBase + m;
            const float val = acc[v] * norm[row];
            hw[(size_t)row * DOUT + colBase + l16] = val;
        }
        __syncthreads();
    }
}

// ------------------------------------------------------------ edge scatter
// One thread handles 4 consecutive features of one edge: float4 gather from
// hw[src] (L2-resident, coalesced within the wave) + 4 atomic f32 adds.
__global__ __launch_bounds__(256)
void gcn_scatter_kernel(const float* __restrict__ hw,
                        const int* __restrict__ src,
                        const int* __restrict__ dst,
                        float* __restrict__ out,
                        int total) {                 // total = E * 32
    int gid = blockIdx.x * 256 + threadIdx.x;
    if (gid >= total) return;
    const int e = gid >> 5;                          // edge id
    const int f = (gid & 31) * 4;                    // feature base
    const int s = src[e];
    const int d = dst[e];
    const float4 v = *(const float4*)(hw + (size_t)s * DOUT + f);
    float* o = out + (size_t)d * DOUT + f;
    atomicAdd(o + 0, v.x);
    atomicAdd(o + 1, v.y);
    atomicAdd(o + 2, v.z);
    atomicAdd(o + 3, v.w);
}

// ---------------------------------------------------------------- finalize
__global__ __launch_bounds__(256)
void gcn_finalize_kernel(float* __restrict__ out,
                         const float* __restrict__ norm,
                         const float* __restrict__ bias,
                         int total) {                // total = N * DOUT
    int gid = blockIdx.x * 256 + threadIdx.x;
    if (gid >= total) return;
    const int row = gid >> 7;                        // DOUT == 128
    const int f   = gid & 127;
    const float v = out[gid] * norm[row] + bias[f];
    out[gid] = v > 0.f ? v : 0.f;
}

// ---------------------------------------------------------------------------
extern "C" void kernel_launch(void* const* d_in, const int* in_sizes, int n_in,
                              void* d_out, int out_size, void* d_ws, size_t ws_size,
                              hipStream_t stream) {
    const float* h    = (const float*)d_in[0];   // [N, 128]
    const float* w    = (const float*)d_in[1];   // [128, 128]
    const float* bias = (const float*)d_in[2];   // [128]
    const float* norm = (const float*)d_in[3];   // [N, 1]
    const int*   src  = (const int*)d_in[4];     // [E]
    const int*   dst  = (const int*)d_in[5];     // [E]

    const int N = in_sizes[3];                   // 50000
    const int E = in_sizes[4];                   // 800000

    float* out = (float*)d_out;                  // [N, 128]
    float* hw  = (float*)d_ws;                   // scratch: N*128 floats

    // 1) zero accumulation buffer (harness poisons d_out)
    {
        const int n4 = (N * DOUT) / 4;
        gcn_zero_kernel<<<(n4 + 255) / 256, 256, 0, stream>>>((float4*)out, n4);
    }
    // 2) WMMA projection + pre-norm  (625 blocks for N = 50000)
    {
        const int blocks = N / (16 * ROW_TILES_PER_BLOCK);
        gcn_gemm_norm_kernel<<<blocks, 256, 0, stream>>>(h, w, norm, hw);
    }
    // 3) neighbor aggregation via atomics into d_out
    {
        const int total = E * 32;
        gcn_scatter_kernel<<<(total + 255) / 256, 256, 0, stream>>>(hw, src, dst, out, total);
    }
    // 4) post-norm + bias + relu
    {
        const int total = N * DOUT;
        gcn_finalize_kernel<<<(total + 255) / 256, 256, 0, stream>>>(out, norm, bias, total);
    }
}